// OpenPanguMoEGate_9620726743827
// MI455X (gfx1250) — compile-verified
//
#include <hip/hip_runtime.h>

typedef float v2f __attribute__((ext_vector_type(2)));
typedef float v4f __attribute__((ext_vector_type(4)));
typedef float v8f __attribute__((ext_vector_type(8)));

#define HDIM 4096
#define NEXP 64
#define TOPK 8
#define NW 2                      // waves per block
#define TOK_PER_WAVE 32           // two 16-token WMMA tiles per wave
#define BLK_TOK (NW * TOK_PER_WAVE)  // 64 tokens per block
#define LDS_PITCH 65              // 64 + 1 pad -> conflict-free per-token scans

__device__ __forceinline__ v8f wmma_f32_k4(v2f a, v2f b, v8f c) {
  // D = A(16x4 f32) * B(4x16 f32) + C(16x16 f32)
  return __builtin_amdgcn_wmma_f32_16x16x4_f32(
      /*neg_a=*/false, a, /*neg_b=*/false, b,
      /*c_mod=*/(short)0, c, /*reuse_a=*/false, /*reuse_b=*/false);
}

__global__ __launch_bounds__(NW * 32)
void moe_gate_wmma_kernel(const float* __restrict__ x,
                          const float* __restrict__ w,
                          float* __restrict__ out, int T) {
  __shared__ float scores[BLK_TOK * LDS_PITCH];

  const int tid  = threadIdx.x;
  const int wave = tid >> 5;
  const int lane = tid & 31;
  const int row  = lane & 15;   // token row (A) / expert row (B)
  const int hi   = lane >> 4;   // half-wave: selects K-slot pair

  // First token of this wave's tile0; tile1 = tok0 + 16
  const int tok0 = blockIdx.x * BLK_TOK + wave * TOK_PER_WAVE;

  // A pointers: lane (row,hi) streams x[tok + row][k + 4*hi ...], step 8 floats
  const float* a0p = x + (size_t)(tok0 + row) * HDIM + hi * 4;
  const float* a1p = x + (size_t)(tok0 + 16 + row) * HDIM + hi * 4;

  v8f acc0[4], acc1[4];
#pragma unroll
  for (int n = 0; n < 4; ++n) { acc0[n] = (v8f)0.0f; acc1[n] = (v8f)0.0f; }

  for (int k = 0; k < HDIM; k += 8) {
    const v4f a0 = *(const v4f*)(a0p + k);
    const v4f a1 = *(const v4f*)(a1p + k);
    const v2f a0lo = {a0.x, a0.y}, a0hi = {a0.z, a0.w};
    const v2f a1lo = {a1.x, a1.y}, a1hi = {a1.z, a1.w};
#pragma unroll
    for (int n = 0; n < 4; ++n) {
      // B fragment: lanes 0-15 -> weight[n*16+row][k..k+3], lanes 16-31 -> [k+4..k+7]
      const v4f b = *(const v4f*)(w + (size_t)(n * 16 + row) * HDIM + hi * 4 + k);
      const v2f blo = {b.x, b.y}, bhi = {b.z, b.w};
      acc0[n] = wmma_f32_k4(a0lo, blo, acc0[n]);   // covers k, k+1, k+4, k+5
      acc0[n] = wmma_f32_k4(a0hi, bhi, acc0[n]);   // covers k+2, k+3, k+6, k+7
      acc1[n] = wmma_f32_k4(a1lo, blo, acc1[n]);
      acc1[n] = wmma_f32_k4(a1hi, bhi, acc1[n]);
    }
  }

  // Sigmoid + scatter to LDS.
  // C/D layout: VGPR v, lane -> token = v + 8*hi (+tile offset), expert = n*16 + row.
#pragma unroll
  for (int n = 0; n < 4; ++n) {
#pragma unroll
    for (int v = 0; v < 8; ++v) {
      const int tloc = wave * TOK_PER_WAVE + v + 8 * hi;
      const int e    = n * 16 + row;
      scores[tloc * LDS_PITCH + e]        = 1.0f / (1.0f + __expf(-acc0[n][v]));
      scores[(tloc + 16) * LDS_PITCH + e] = 1.0f / (1.0f + __expf(-acc1[n][v]));
    }
  }
  __syncthreads();

  // Top-8 selection: one thread per token (64 threads, 64 tokens).
  float* srow = &scores[tid * LDS_PITCH];
  const int gtok = blockIdx.x * BLK_TOK + tid;

  float wv[TOPK];
  int   wi[TOPK];
  float sum = 0.0f;
#pragma unroll
  for (int p = 0; p < TOPK; ++p) {
    float best = -1.0f;  // sigmoid outputs are in (0,1)
    int   bi   = 0;
    for (int e = 0; e < NEXP; ++e) {
      const float s = srow[e];
      if (s > best) { best = s; bi = e; }
    }
    srow[bi] = -2.0f;  // remove from further passes
    wv[p] = best;
    wi[p] = bi;
    sum  += best;
  }
  const float scale = 2.5f / (sum + 1e-20f);

  float* oidx = out + (size_t)gtok * TOPK;                    // topk_idx region
  float* owgt = out + (size_t)T * TOPK + (size_t)gtok * TOPK; // topk_weight region
#pragma unroll
  for (int p = 0; p < TOPK; ++p) {
    oidx[p] = (float)wi[p];
    owgt[p] = wv[p] * scale;
  }
}

extern "C" void kernel_launch(void* const* d_in, const int* in_sizes, int n_in,
                              void* d_out, int out_size, void* d_ws, size_t ws_size,
                              hipStream_t stream) {
  const float* x = (const float*)d_in[0];   // hidden_states [4,4096,4096] f32
  const float* w = (const float*)d_in[1];   // weight [64,4096] f32
  float* out = (float*)d_out;

  const int T = in_sizes[0] / HDIM;         // 16384 tokens
  const int blocks = T / BLK_TOK;           // 256 blocks

  moe_gate_wmma_kernel<<<blocks, NW * 32, 0, stream>>>(x, w, out, T);
}